// GraphSpectralFilterLayer_82102594830726
// MI455X (gfx1250) — compile-verified
//
#include <hip/hip_runtime.h>
#include <hip/hip_bf16.h>

// Problem constants (from the reference setup_inputs()).
#define NROWS   8192     // N
#define IN_F    512
#define OUT_F   64
#define NCONC   2        // C
#define RROWS   (NCONC * NROWS)   // 16384
#define ROW_LEN NROWS    // attention row length = 8192
#define KTOP    32
#define TPB     256      // threads per block for topk kernel
#define ELEMS   (ROW_LEN / TPB)   // 32 private elements per thread

typedef __attribute__((ext_vector_type(2))) float v2f;
typedef __attribute__((ext_vector_type(8))) float v8f;

// ---------------------------------------------------------------------------
// Kernel A: h = x @ W^T  via V_WMMA_F32_16X16X4_F32.
// One wave computes one 16x16 tile of h. Block = 128 threads = 4 waves
// covering all 4 N-tiles (OUT_F=64) of one 16-row M-tile. Grid = 512.
// Fragment layouts per CDNA5 ISA 7.12.2.
// ---------------------------------------------------------------------------
__global__ __launch_bounds__(128)
void linear_wmma_kernel(const float* __restrict__ x,
                        const float* __restrict__ W,
                        float* __restrict__ h) {
    const int lane = threadIdx.x & 31;
    const int wave = threadIdx.x >> 5;          // 0..3 -> N tile
    const int tm   = blockIdx.x * 16;           // M tile base
    const int tn   = wave * 16;                 // N tile base
    const int m    = lane & 15;
    const int kb   = (lane < 16) ? 0 : 2;       // K sub-offset for this half-wave

    const float* xr = x + (size_t)(tm + m) * IN_F;   // A row (M = m)
    const float* wr = W + (size_t)(tn + m) * IN_F;   // B col n=m of W^T == W row tn+m

    v8f acc = {};
    for (int k0 = 0; k0 < IN_F; k0 += 4) {
        v2f a, b;
        a.x = xr[k0 + kb];
        a.y = xr[k0 + kb + 1];
        b.x = wr[k0 + kb];
        b.y = wr[k0 + kb + 1];
        acc = __builtin_amdgcn_wmma_f32_16x16x4_f32(
                  /*neg_a=*/false, a, /*neg_b=*/false, b,
                  /*c_mod=*/(short)0, acc, /*reuse_a=*/false, /*reuse_b=*/false);
    }

    float* hr = h + (size_t)tm * OUT_F + tn;
    #pragma unroll
    for (int v = 0; v < 8; ++v) {
        const int mm = v + ((lane < 16) ? 0 : 8);
        hr[(size_t)mm * OUT_F + (lane & 15)] = acc[v];
    }
}

// ---------------------------------------------------------------------------
// Order-preserving f32 <-> u32 key transform (descending top-k == largest keys)
// ---------------------------------------------------------------------------
__device__ __forceinline__ unsigned float_to_key(float f) {
    unsigned u = __float_as_uint(f);
    return (u >> 31) ? ~u : (u | 0x80000000u);
}
__device__ __forceinline__ float key_to_float(unsigned k) {
    unsigned u = (k & 0x80000000u) ? (k ^ 0x80000000u) : ~k;
    return __uint_as_float(u);
}

// ---------------------------------------------------------------------------
// Kernel B: per-row exact top-32 via 4-level MSB radix select (256-bin LDS
// histogram + suffix scan per level), then softmax over the 32 winners,
// dense zero-fill + scatter of the att row, and (idx,weight) lists to ws.
// One 256-thread block per row; each thread holds 32 strided keys in regs.
// Streaming traffic (araw read / att zero-fill) uses non-temporal hints.
// ---------------------------------------------------------------------------
__global__ __launch_bounds__(TPB)
void topk_softmax_kernel(const float* __restrict__ araw,
                         float* __restrict__ att,
                         int*   __restrict__ tidx,
                         float* __restrict__ tw) {
    const int row = blockIdx.x;
    const int t   = threadIdx.x;
    const float* rp = araw + (size_t)row * ROW_LEN;

    // Load private slice (coalesced, non-temporal: single-use stream).
    unsigned kv[ELEMS];
    #pragma unroll
    for (int i = 0; i < ELEMS; ++i) {
        float f = __builtin_nontemporal_load(rp + t + i * TPB);
        kv[i] = float_to_key(f);
    }

    __shared__ unsigned hist[256];
    __shared__ unsigned scan[256];
    __shared__ unsigned sh_prefix;
    __shared__ int      sh_remaining;
    __shared__ float    topv[KTOP];
    __shared__ int      topc[KTOP];
    __shared__ int      sh_pos;
    __shared__ float    sh_max, sh_sum;

    if (t == 0) { sh_prefix = 0u; sh_remaining = KTOP; sh_pos = 0; }
    __syncthreads();

    // ---- 4-level radix select: find exact 32nd-largest key ----
    #pragma unroll
    for (int l = 0; l < 4; ++l) {
        const int shift = 24 - 8 * l;
        hist[t] = 0u;
        __syncthreads();
        const unsigned p = sh_prefix;

        #pragma unroll
        for (int i = 0; i < ELEMS; ++i) {
            const unsigned key = kv[i];
            if ((unsigned)(((unsigned long long)key) >> (shift + 8)) == p)
                atomicAdd(&hist[(key >> shift) & 0xFFu], 1u);
        }
        __syncthreads();

        // inclusive suffix sum: scan[t] = sum_{j>=t} hist[j]
        scan[t] = hist[t];
        __syncthreads();
        #pragma unroll
        for (int off = 1; off < 256; off <<= 1) {
            const unsigned add = (t + off < 256) ? scan[t + off] : 0u;
            __syncthreads();
            scan[t] += add;
            __syncthreads();
        }

        const int rem = sh_remaining;
        const unsigned above = (t < 255) ? scan[t + 1] : 0u;   // strictly above digit t
        if ((int)above < rem && (int)(above + hist[t]) >= rem) {
            sh_prefix    = (p << 8) | (unsigned)t;
            sh_remaining = rem - (int)above;                   // ties to take at digit t
        }
        __syncthreads();
    }

    const unsigned T = sh_prefix;   // exact threshold key (32nd largest)

    // ---- collect: all keys > T, then exactly `remaining` ties == T ----
    #pragma unroll
    for (int i = 0; i < ELEMS; ++i) {
        const unsigned key = kv[i];
        if (key > T) {
            const int s = atomicAdd(&sh_pos, 1);
            topv[s] = key_to_float(key);
            topc[s] = t + i * TPB;
        }
    }
    __syncthreads();
    #pragma unroll
    for (int i = 0; i < ELEMS; ++i) {
        const unsigned key = kv[i];
        if (key == T) {
            const int s = atomicAdd(&sh_pos, 1);
            if (s < KTOP) {
                topv[s] = key_to_float(key);
                topc[s] = t + i * TPB;
            }
        }
    }
    __syncthreads();

    // ---- softmax over the 32 winners ----
    if (t == 0) {
        float mx = topv[0];
        #pragma unroll
        for (int i = 1; i < KTOP; ++i) mx = fmaxf(mx, topv[i]);
        float s = 0.0f;
        #pragma unroll
        for (int i = 0; i < KTOP; ++i) s += expf(topv[i] - mx);
        sh_max = mx;
        sh_sum = s;
    }
    __syncthreads();

    // ---- dense att row: zeros (masked entries are exactly 0 in reference),
    //      non-temporal (single-use 537 MB stream), then scatter winners ----
    float* ap = att + (size_t)row * ROW_LEN;
    #pragma unroll
    for (int i = 0; i < ELEMS; ++i)
        __builtin_nontemporal_store(0.0f, ap + t + i * TPB);
    __threadfence();
    __syncthreads();

    if (t < KTOP) {
        const float p = expf(topv[t] - sh_max) / sh_sum;
        ap[topc[t]] = p;
        tidx[(size_t)row * KTOP + t] = topc[t];
        tw  [(size_t)row * KTOP + t] = p;
    }
}

// ---------------------------------------------------------------------------
// Kernel C: h_prime = att @ h using the k-sparse lists, fused with the
// [C,N,F] -> [N, C*F] concat permutation. One 64-thread block per sparse row;
// h (2 MB) is L2-resident so the gathers are cheap.
// ---------------------------------------------------------------------------
__global__ __launch_bounds__(OUT_F)
void spmm_concat_kernel(const float* __restrict__ h,
                        const int*   __restrict__ tidx,
                        const float* __restrict__ tw,
                        float* __restrict__ out) {
    const int r = blockIdx.x;          // 0..RROWS-1
    const int f = threadIdx.x;         // 0..63

    __shared__ int   sidx[KTOP];
    __shared__ float sw[KTOP];
    if (f < KTOP) {
        sidx[f] = tidx[(size_t)r * KTOP + f];
        sw[f]   = tw  [(size_t)r * KTOP + f];
    }
    __syncthreads();

    float acc = 0.0f;
    #pragma unroll
    for (int j = 0; j < KTOP; ++j) {
        acc += sw[j] * h[(size_t)sidx[j] * OUT_F + f];
    }

    const int c = r / NROWS;
    const int n = r - c * NROWS;
    out[(size_t)n * (NCONC * OUT_F) + c * OUT_F + f] = acc;
}

// ---------------------------------------------------------------------------
extern "C" void kernel_launch(void* const* d_in, const int* in_sizes, int n_in,
                              void* d_out, int out_size, void* d_ws, size_t ws_size,
                              hipStream_t stream) {
    const float* x    = (const float*)d_in[0];   // [8192, 512]
    const float* W    = (const float*)d_in[1];   // [64, 512]
    const float* araw = (const float*)d_in[2];   // [16384, 8192]
    // d_in[3] = k (==32), baked into KTOP.

    // Output tuple (out, att) flat-concatenated.
    float* out = (float*)d_out;                          // [8192, 128]
    float* att = out + (size_t)NROWS * (NCONC * OUT_F);  // [16384, 8192]

    // Workspace: h (2 MB) + topk index (2 MB) + topk weight (2 MB).
    float* h    = (float*)d_ws;                          // [8192, 64]
    int*   tidx = (int*)(h + (size_t)NROWS * OUT_F);     // [16384, 32]
    float* tw   = (float*)(tidx + (size_t)RROWS * KTOP); // [16384, 32]

    // A: dense linear via f32 WMMA
    linear_wmma_kernel<<<NROWS / 16, 128, 0, stream>>>(x, W, h);

    // B: radix-select top-32 + softmax + dense att write (dominant, memory-bound)
    topk_softmax_kernel<<<RROWS, TPB, 0, stream>>>(araw, att, tidx, tw);

    // C: k-sparse spmm + concat permutation
    spmm_concat_kernel<<<RROWS, OUT_F, 0, stream>>>(h, tidx, tw, out);
}